// BaselineArch_16776142258589
// MI455X (gfx1250) — compile-verified
//
#include <hip/hip_runtime.h>

#define NN 50000
#define NE 800000
#define D 64
#define MTILES (NN / 16)   // 3125, exact

typedef float v2f __attribute__((ext_vector_type(2)));
typedef float v8f __attribute__((ext_vector_type(8)));

// ---------------------------------------------------------------------------
// Zero the workspace accumulators (float4 stores, 16B per thread)
// ---------------------------------------------------------------------------
__global__ void zero_f4(float* __restrict__ p, int n4) {
    int i = blockIdx.x * blockDim.x + threadIdx.x;
    if (i < n4) {
        float4 z = {0.f, 0.f, 0.f, 0.f};
        ((float4*)p)[i] = z;
    }
}

// ---------------------------------------------------------------------------
// Weighted gather/scatter-add:  acc[dst] += w[e] * Feat[src]
// 16 consecutive threads cover one edge's 64-float row with float4 loads
// (full 256B cachelines, L2-resident since Feat is only 12.8MB).
// ---------------------------------------------------------------------------
__global__ void edge_scatter(const float* __restrict__ Feat,
                             const long long* __restrict__ ei,   // [2, NE] int64
                             const float* __restrict__ ew,
                             float* __restrict__ acc) {
    long long gid = (long long)blockIdx.x * blockDim.x + threadIdx.x;
    int e = (int)(gid >> 4);
    int q = (int)(gid & 15);
    if (e >= NE) return;
    int src = (int)ei[e];
    int dst = (int)ei[NE + e];
    float w = ew[e];
    float4 v = *(const float4*)(Feat + (size_t)src * D + q * 4);
    float* out = acc + (size_t)dst * D + q * 4;
    atomicAdd(out + 0, w * v.x);
    atomicAdd(out + 1, w * v.y);
    atomicAdd(out + 2, w * v.z);
    atomicAdd(out + 3, w * v.w);
}

// ---------------------------------------------------------------------------
// Out = [relu](A @ W + bias), A: M x 64, W: 64 x 64, exact f32 via
// V_WMMA_F32_16X16X4_F32. One wave per 16x16 output tile, 16 WMMA ops (K=64).
//
// Fragment layouts (ISA 7.12.2, wave32):
//   A 16x4:  lane L holds row m=L%16; VGPR pair = K {k0+2*(L/16), +1}
//   B 4x16:  lane L holds col n=L%16; VGPR pair = K {k0+2*(L/16), +1}
//   C/D 16x16: VGPR i, lane L -> row i + 8*(L/16), col L%16
// ---------------------------------------------------------------------------
template <bool RELU>
__global__ void gemm64_wmma(const float* __restrict__ A,
                            const float* __restrict__ W,
                            const float* __restrict__ bias,
                            float* __restrict__ Out,
                            int mtiles) {
    int wave = (blockIdx.x * blockDim.x + threadIdx.x) >> 5;
    int lane = threadIdx.x & 31;
    int nt = wave & 3;          // 4 column tiles (N=64)
    int mt = wave >> 2;
    if (mt >= mtiles) return;   // wave-uniform: EXEC stays all-ones for WMMA

    int l16 = lane & 15;
    int hi  = (lane >> 4) << 1;         // 0 for lanes 0-15, 2 for lanes 16-31
    int m   = (mt << 4) + l16;          // A row owned by this lane
    int n   = (nt << 4) + l16;          // output column owned by this lane

    const float* arow = A + (size_t)m * D;
    v8f c = {0.f, 0.f, 0.f, 0.f, 0.f, 0.f, 0.f, 0.f};

#pragma unroll
    for (int k0 = 0; k0 < D; k0 += 4) {
        v2f a = *(const v2f*)(arow + k0 + hi);          // A[m][k0+hi .. +1]
        v2f b;
        b.x = W[(size_t)(k0 + hi) * D + n];             // W[k][n]
        b.y = W[(size_t)(k0 + hi + 1) * D + n];
        c = __builtin_amdgcn_wmma_f32_16x16x4_f32(
                /*neg_a=*/false, a, /*neg_b=*/false, b,
                /*c_mod=*/(short)0, c, /*reuse_a=*/false, /*reuse_b=*/false);
    }

    float bv = bias[n];
    int rbase = (mt << 4) + ((lane >> 4) << 3);
#pragma unroll
    for (int i = 0; i < 8; ++i) {
        float v = c[i] + bv;
        if (RELU) v = fmaxf(v, 0.f);
        Out[(size_t)(rbase + i) * D + n] = v;
    }
}

// ---------------------------------------------------------------------------
// Inputs (setup_inputs order): X, edge_weight, W1, b1, W2, b2, edge_index
// ---------------------------------------------------------------------------
extern "C" void kernel_launch(void* const* d_in, const int* in_sizes, int n_in,
                              void* d_out, int out_size, void* d_ws, size_t ws_size,
                              hipStream_t stream) {
    const float*     X  = (const float*)d_in[0];
    const float*     ew = (const float*)d_in[1];
    const float*     W1 = (const float*)d_in[2];
    const float*     b1 = (const float*)d_in[3];
    const float*     W2 = (const float*)d_in[4];
    const float*     b2 = (const float*)d_in[5];
    const long long* ei = (const long long*)d_in[6];
    float* out = (float*)d_out;

    float* A1 = (float*)d_ws;          // 50000*64 f32  (agg(X))
    float* H  = A1 + (size_t)NN * D;   // 50000*64 f32  (relu layer-1 out)
    float* A2 = H  + (size_t)NN * D;   // 50000*64 f32  (agg(H))

    // zero all three ws regions (A1/A2 must be zeroed every call; H harmless)
    {
        int n4 = 3 * NN * D / 4;                       // 2,400,000 float4
        zero_f4<<<(n4 + 255) / 256, 256, 0, stream>>>((float*)d_ws, n4);
    }

    // layer 1 aggregation: A1 = scatter_add(ew * X[src] -> dst)
    {
        long long thr = (long long)NE * 16;            // 12.8M
        edge_scatter<<<(int)((thr + 255) / 256), 256, 0, stream>>>(X, ei, ew, A1);
    }

    // H = relu(A1 @ W1 + b1)
    {
        int waves = MTILES * 4;                        // 12500
        int blocks = (waves + 7) / 8;                  // 8 waves / 256-thr block
        gemm64_wmma<true><<<blocks, 256, 0, stream>>>(A1, W1, b1, H, MTILES);
    }

    // layer 2 aggregation: A2 = scatter_add(ew * H[src] -> dst)
    {
        long long thr = (long long)NE * 16;
        edge_scatter<<<(int)((thr + 255) / 256), 256, 0, stream>>>(H, ei, ew, A2);
    }

    // out = A2 @ W2 + b2
    {
        int waves = MTILES * 4;
        int blocks = (waves + 7) / 8;
        gemm64_wmma<false><<<blocks, 256, 0, stream>>>(A2, W2, b2, out, MTILES);
    }
}